// DirSageConv_36567351558755
// MI455X (gfx1250) — compile-verified
//
#include <hip/hip_runtime.h>

typedef __attribute__((ext_vector_type(2))) float v2f;
typedef __attribute__((ext_vector_type(8))) float v8f;
typedef __attribute__((ext_vector_type(4))) unsigned int u32x4;
typedef __attribute__((ext_vector_type(8))) int i32x8;
typedef __attribute__((ext_vector_type(4))) int i32x4;

#define D 64

#if __has_builtin(__builtin_amdgcn_tensor_load_to_lds)
#define HAVE_TDM 1
#endif

// ---------------------------------------------------------------- zero
__global__ void zero_f32(float* __restrict__ p, long n) {
  long i = (long)blockIdx.x * blockDim.x + threadIdx.x;
  long stride = (long)gridDim.x * blockDim.x;
  for (; i < n; i += stride) p[i] = 0.0f;
}

// ------------------------------------------------------------- scatter
// One wave32 per edge. Lane l handles columns {2l, 2l+1} (float2).
// Aggregate sum buffers (51 MB) + x (25.6 MB) are L2-resident on MI455X
// (192 MB L2), so the f32 atomic RMWs stay in L2, not HBM.
__global__ __launch_bounds__(256)
void scatter_kernel(const float* __restrict__ x,
                    const long long* __restrict__ src,
                    const long long* __restrict__ dst,
                    float* __restrict__ agg_in,
                    float* __restrict__ agg_out,
                    float* __restrict__ cnt_in,
                    float* __restrict__ cnt_out,
                    int E) {
  int wave = (int)((blockIdx.x * blockDim.x + threadIdx.x) >> 5);
  int lane = threadIdx.x & 31;
  if (wave >= E) return;
  int s = (int)src[wave];
  int d = (int)dst[wave];
  const float2 xs = *(const float2*)(x + (long)s * D + 2 * lane);
  const float2 xd = *(const float2*)(x + (long)d * D + 2 * lane);
  float* ai = agg_in + (long)d * D + 2 * lane;
  float* ao = agg_out + (long)s * D + 2 * lane;
  atomicAdd(ai + 0, xs.x);
  atomicAdd(ai + 1, xs.y);
  atomicAdd(ao + 0, xd.x);
  atomicAdd(ao + 1, xd.y);
  if (lane == 0) {
    atomicAdd(cnt_in + d, 1.0f);
    atomicAdd(cnt_out + s, 1.0f);
  }
}

// --------------------------------------------------- TDM weight staging
// 1-D fp32 tile: n contiguous elements global -> LDS. D# per ISA 08 §8:
// g0 = {count=1, lds_addr, global_addr[31:0], global_addr[56:32]|type=2<<30}
// g1 = {data_size=4B, tensor_dim0=n, tensor_dim1=1, tile_dim0=n, tile_dim1=1,
//       tensor_dim0_stride=n}; groups 2/3 zero (<=2-D tensor).
// This toolchain exposes the 6-arg builtin form (extra i32x8 before cpol).
__device__ __forceinline__ void tdm_load_1d(unsigned lds_off, const float* gptr,
                                            unsigned n) {
#ifdef HAVE_TDM
  unsigned long long ga = (unsigned long long)(const void*)gptr;
  u32x4 g0 = { 1u,
               lds_off,
               (unsigned)(ga & 0xFFFFFFFFu),
               (unsigned)((ga >> 32) & 0x01FFFFFFu) | 0x80000000u };
  i32x8 g1 = { (int)(2u << 16),                     // data_size = 2 -> 4 bytes
               (int)((n & 0xFFFFu) << 16),          // tensor_dim0 lo16
               (int)((n >> 16) & 0xFFFFu) | (1 << 16), // dim0 hi16 | tensor_dim1=1
               (int)((n & 0xFFFFu) << 16),          // tile_dim0
               1,                                   // tile_dim1 = 1
               (int)n,                              // tensor_dim0_stride lo32
               0, 0 };
  i32x4 gz4 = { 0, 0, 0, 0 };
  i32x8 gz8 = { 0, 0, 0, 0, 0, 0, 0, 0 };
  __builtin_amdgcn_tensor_load_to_lds(g0, g1, gz4, gz4, gz8, 0);
#else
  (void)lds_off; (void)gptr; (void)n;
#endif
}

// ------------------------------------------------------------ finalize
// Fused single GEMM with K=192: out = A' @ W' + b,
//   A' = [x | agg_in*(0.5/deg_in) | agg_out*(0.5/deg_out)]  (built in regs)
//   W' = [W_self ; W_s2d ; W_d2s]  staged in 48 KB LDS via TDM.
// Full fp32 via V_WMMA_F32_16X16X4_F32 (2.5 GFLOP total: bandwidth-bound,
// so no precision downcast needed). Three branch-free unrolled region loops
// let the scheduler batch all 16 global_load_b64 per region before the WMMAs.
template <bool SCALE>
__device__ __forceinline__ void gemm_region(const float* __restrict__ rowp,
                                            float sc, const float* Wr,
                                            int half, int lrow,
                                            v8f& c0, v8f& c1, v8f& c2, v8f& c3) {
#pragma unroll
  for (int j = 0; j < 16; ++j) {
    const int kk = 4 * j + 2 * half;          // K pair this half-wave supplies
    float2 t = *(const float2*)(rowp + kk);
    v2f a;
    if (SCALE) { a.x = t.x * sc; a.y = t.y * sc; }
    else       { a.x = t.x;      a.y = t.y; }
    const float* w0 = Wr + kk * 64;           // lane stride 1 DWORD: bank-clean
    const float* w1 = w0 + 64;
    v2f b;
    b.x = w0[lrow +  0]; b.y = w1[lrow +  0];
    c0 = __builtin_amdgcn_wmma_f32_16x16x4_f32(false, a, false, b, (short)0, c0, false, false);
    b.x = w0[lrow + 16]; b.y = w1[lrow + 16];
    c1 = __builtin_amdgcn_wmma_f32_16x16x4_f32(false, a, false, b, (short)0, c1, false, false);
    b.x = w0[lrow + 32]; b.y = w1[lrow + 32];
    c2 = __builtin_amdgcn_wmma_f32_16x16x4_f32(false, a, false, b, (short)0, c2, false, false);
    b.x = w0[lrow + 48]; b.y = w1[lrow + 48];
    c3 = __builtin_amdgcn_wmma_f32_16x16x4_f32(false, a, false, b, (short)0, c3, false, false);
  }
}

__global__ __launch_bounds__(256)
void finalize_kernel(const float* __restrict__ x,
                     const float* __restrict__ W_self,
                     const float* __restrict__ W_s2d,
                     const float* __restrict__ W_d2s,
                     const float* __restrict__ b_self,
                     const float* __restrict__ agg_in,
                     const float* __restrict__ agg_out,
                     const float* __restrict__ cnt_in,
                     const float* __restrict__ cnt_out,
                     float* __restrict__ out,
                     int N) {
  __shared__ float Wp[192 * 64];  // 48 KB stacked weights [W_self; W_s2d; W_d2s]

#ifdef HAVE_TDM
  if (threadIdx.x < 32) {  // wave 0 only: TDM issues once per executing wave
    unsigned lds0 = (unsigned)(unsigned long long)(uintptr_t)&Wp[0];
    tdm_load_1d(lds0 +     0u, W_self, 4096u);
    tdm_load_1d(lds0 + 16384u, W_s2d,  4096u);
    tdm_load_1d(lds0 + 32768u, W_d2s,  4096u);
#if __has_builtin(__builtin_amdgcn_s_wait_tensorcnt)
    __builtin_amdgcn_s_wait_tensorcnt(0);
#endif
  }
#else
  for (int i = threadIdx.x; i < 64 * 64; i += blockDim.x) {
    Wp[i]            = W_self[i];
    Wp[64 * 64 + i]  = W_s2d[i];
    Wp[128 * 64 + i] = W_d2s[i];
  }
#endif
  __syncthreads();

  const int wave = threadIdx.x >> 5;
  const int lane = threadIdx.x & 31;
  const int r0 = blockIdx.x * 128 + wave * 16;  // 16-row tile per wave
  if (r0 >= N) return;  // N % 16 == 0; wave-uniform guard, EXEC all-1s in WMMA

  const int half = lane >> 4;   // 0: lanes 0-15 (K+0,K+1), 1: lanes 16-31 (K+2,K+3)
  const int lrow = lane & 15;   // A row within tile / B column within n-tile
  const long arow = (long)r0 + lrow;

  const float sc_in  = 0.5f / fmaxf(cnt_in[arow], 1.0f);
  const float sc_out = 0.5f / fmaxf(cnt_out[arow], 1.0f);

  v8f c0 = {}, c1 = {}, c2 = {}, c3 = {};

  gemm_region<false>(x       + arow * D, 1.0f,   &Wp[0],        half, lrow, c0, c1, c2, c3);
  gemm_region<true >(agg_in  + arow * D, sc_in,  &Wp[64 * 64],  half, lrow, c0, c1, c2, c3);
  gemm_region<true >(agg_out + arow * D, sc_out, &Wp[128 * 64], half, lrow, c0, c1, c2, c3);

  // 16x16 f32 C/D layout: VGPR v -> M = v + 8*half, N = lrow.
#pragma unroll
  for (int v = 0; v < 8; ++v) {
    const int m = v + 8 * half;
    float* od = out + ((long)r0 + m) * D + lrow;
    od[ 0] = c0[v] + b_self[lrow +  0];
    od[16] = c1[v] + b_self[lrow + 16];
    od[32] = c2[v] + b_self[lrow + 32];
    od[48] = c3[v] + b_self[lrow + 48];
  }
}

// -------------------------------------------------------------- launch
extern "C" void kernel_launch(void* const* d_in, const int* in_sizes, int n_in,
                              void* d_out, int out_size, void* d_ws, size_t ws_size,
                              hipStream_t stream) {
  const float*     x      = (const float*)d_in[0];
  const long long* eidx   = (const long long*)d_in[1];  // int64 per reference
  const float*     W_s2d  = (const float*)d_in[2];
  const float*     W_d2s  = (const float*)d_in[3];
  const float*     W_self = (const float*)d_in[4];
  const float*     b_self = (const float*)d_in[5];
  float*           out    = (float*)d_out;

  const int N = in_sizes[0] / D;   // 100000
  const int E = in_sizes[1] / 2;   // 1000000
  const long long* src = eidx;     // edge_index[0]
  const long long* dst = eidx + E; // edge_index[1]

  // Workspace: agg_in_sum[N*64] | agg_out_sum[N*64] | cnt_in[N] | cnt_out[N]
  float* ws      = (float*)d_ws;
  float* agg_in  = ws;
  float* agg_out = ws + (size_t)N * D;
  float* cnt_in  = ws + (size_t)2 * N * D;
  float* cnt_out = cnt_in + N;

  const long zn = (long)2 * N * D + 2 * N;
  zero_f32<<<2048, 256, 0, stream>>>(ws, zn);

  const int edge_blocks = (E + 7) / 8;  // 8 waves (256 threads) per block
  scatter_kernel<<<edge_blocks, 256, 0, stream>>>(x, src, dst, agg_in, agg_out,
                                                  cnt_in, cnt_out, E);

  finalize_kernel<<<(N + 127) / 128, 256, 0, stream>>>(x, W_self, W_s2d, W_d2s,
                                                       b_self, agg_in, agg_out,
                                                       cnt_in, cnt_out, out, N);
}